// MorphLayer_24927990186537
// MI455X (gfx1250) — compile-verified
//
#include <hip/hip_runtime.h>
#include <math.h>

typedef unsigned int u32;
typedef u32 v4u __attribute__((ext_vector_type(4)));
typedef int  v8i __attribute__((ext_vector_type(8)));
typedef int  v4i __attribute__((ext_vector_type(4)));

#define B_   16
#define C_   3
#define H_   66
#define W_   66
#define F_   32
#define KK   27      // 3*3*3 patch
#define HO   64
#define WO   64
#define TILE_ROWS 4
#define IN_ROWS   (TILE_ROWS + 2)   // 6 input rows per tile
#define MIN_VAL   1e-12f

__global__ __launch_bounds__(256) void morph_kernel(
    const float* __restrict__ x,
    const float* __restrict__ k1,
    const float* __restrict__ k2,
    const float* __restrict__ bias,
    float* __restrict__ out)
{
    __shared__ float  s_in[C_ * IN_ROWS * W_];  // 1188 floats: 6 rows x 66 cols x 3 ch
    // Interleaved exp'd weights: s_e4[k*16 + f/2] = (e1[k][f], e2[k][f], e1[k][f+1], e2[k][f+1])
    __shared__ float4 s_e4[KK * (F_ / 2)];      // 432 float4 = 6912 B
    __shared__ float  s_bias[F_];

    const int tid = threadIdx.x;
    const int bx  = blockIdx.x;
    const int b   = bx >> 4;                // batch
    const int r0  = (bx & 15) * TILE_ROWS;  // first output row of tile

    // ---- Wave 0: TDM async copy of input tile (3 ch x [6x66 contiguous]) into LDS ----
    if (tid < 32) {
        u32 lds_off = (u32)(unsigned long long)(size_t)(void*)&s_in[0]; // low 32b = LDS byte offset
        const float* gsrc = x + (size_t)b * (C_ * H_ * W_) + (size_t)r0 * W_;
        unsigned long long ga = (unsigned long long)(size_t)(const void*)gsrc;

        v4u g0;
        g0.x = 1u;                                   // count=1, is_restore=0, gather=0
        g0.y = lds_off;                              // lds_addr (bytes)
        g0.z = (u32)(ga & 0xFFFFFFFFu);              // global_addr[31:0]
        g0.w = (u32)((ga >> 32) & 0x01FFFFFFu)       // global_addr[56:32]
             | (2u << 30);                           // type = 2 ("image")

        v8i g1;
        g1[0] = (int)(2u << 16);                     // data_size=2 (4 bytes), mask=0
        g1[1] = (int)((u32)(H_ * W_) << 16);         // tensor_dim0[15:0] @ bits 63:48
        g1[2] = (int)(((u32)(H_ * W_) >> 16)         // tensor_dim0[31:16]
                      | (3u << 16));                 // tensor_dim1[15:0] = 3
        g1[3] = (int)((u32)(IN_ROWS * W_) << 16);    // tile_dim0 = 396 @ bits 127:112
        g1[4] = 3;                                   // tile_dim1 = 3, tile_dim2 = 0
        g1[5] = (int)(H_ * W_);                      // tensor_dim0_stride[31:0] = 4356
        g1[6] = 0;
        g1[7] = 0;

        v4i g2 = {0, 0, 0, 0};
        v4i g3 = {0, 0, 0, 0};
        v8i g4 = {0, 0, 0, 0, 0, 0, 0, 0};
        __builtin_amdgcn_tensor_load_to_lds(g0, g1, g2, g3, g4, 0);
    }

    // ---- All waves: exponentiate + interleave weights into LDS (hides TDM latency) ----
    for (int idx = tid; idx < KK * (F_ / 2); idx += 256) {
        int k  = idx >> 4;          // 0..26
        int fp = idx & 15;          // f-pair index
        int base = k * F_ + 2 * fp;
        float4 w;
        w.x = __expf(k1[base]);
        w.y = __expf(k2[base]);
        w.z = __expf(k1[base + 1]);
        w.w = __expf(k2[base + 1]);
        s_e4[idx] = w;
    }
    if (tid < F_) s_bias[tid] = bias[tid];

    if (tid < 32) __builtin_amdgcn_s_wait_tensorcnt(0);
    __syncthreads();

    // ---- Compute: wave -> (row, col-half); lane -> column (coalesced stores) ----
    const int wave = tid >> 5;
    const int lane = tid & 31;
    const int row  = wave >> 1;                 // 0..3 (local)
    const int j    = ((wave & 1) << 5) + lane;  // 0..63
    const int i    = r0 + row;                  // global output row

    // Load the 27-value patch once into registers; split into +/- rectified parts.
    float xp[KK], xn[KK];
#pragma unroll
    for (int c = 0; c < 3; ++c) {
#pragma unroll
        for (int di = 0; di < 3; ++di) {
#pragma unroll
            for (int dj = 0; dj < 3; ++dj) {
                float v = s_in[(c * IN_ROWS + row + di) * W_ + (j + dj)];
                int k = (c * 3 + di) * 3 + dj;
                xp[k] = fmaxf(v, MIN_VAL);    // exp(log(max(x,eps)))  = max(x,eps)
                xn[k] = fmaxf(-v, MIN_VAL);   // exp(log(max(-x,eps))) = max(-x,eps)
            }
        }
    }

    float* outb = out + (((size_t)b * F_) * HO + i) * WO + j;

    for (int f = 0; f < F_; f += 2) {
        float m11a = 0.0f, m12a = 0.0f, m21a = 0.0f, m22a = 0.0f; // products are > 0
        float m11b = 0.0f, m12b = 0.0f, m21b = 0.0f, m22b = 0.0f;
#pragma unroll
        for (int k = 0; k < KK; ++k) {
            float4 w = s_e4[k * (F_ / 2) + (f >> 1)];  // one ds_load_b128, wave-uniform
            m11a = fmaxf(m11a, xp[k] * w.x);
            m12a = fmaxf(m12a, xp[k] * w.y);
            m21a = fmaxf(m21a, xn[k] * w.x);
            m22a = fmaxf(m22a, xn[k] * w.y);
            m11b = fmaxf(m11b, xp[k] * w.z);
            m12b = fmaxf(m12b, xp[k] * w.w);
            m21b = fmaxf(m21b, xn[k] * w.z);
            m22b = fmaxf(m22b, xn[k] * w.w);
        }
        outb[(size_t)f       * (HO * WO)] = m11a - m12a - m21a + m22a + s_bias[f];
        outb[(size_t)(f + 1) * (HO * WO)] = m11b - m12b - m21b + m22b + s_bias[f + 1];
    }
}

extern "C" void kernel_launch(void* const* d_in, const int* in_sizes, int n_in,
                              void* d_out, int out_size, void* d_ws, size_t ws_size,
                              hipStream_t stream) {
    const float* x    = (const float*)d_in[0];
    const float* k1   = (const float*)d_in[1];
    const float* k2   = (const float*)d_in[2];
    const float* bias = (const float*)d_in[3];
    float* out        = (float*)d_out;
    (void)in_sizes; (void)n_in; (void)out_size; (void)d_ws; (void)ws_size;

    dim3 grid(B_ * (HO / TILE_ROWS));   // 16 batches x 16 row-tiles = 256 blocks
    dim3 block(256);                    // 8 waves (wave32)
    morph_kernel<<<grid, block, 0, stream>>>(x, k1, k2, bias, out);
}